// LEAKYNETWORK_83288005804287
// MI455X (gfx1250) — compile-verified
//
#include <hip/hip_runtime.h>

// ---------------------------------------------------------------------------
// Leaky SNN: spikes = (u < x); cur = spikes @ W.T + b; leaky scan over t=100.
// Phase 1: fp32 WMMA (v_wmma_f32_16x16x4_f32) over 16-row tiles, K=100=25*4.
// Phase 2: per-(batch,o) recurrence with coalesced reads/writes.
// ---------------------------------------------------------------------------

#define DLEN   100
#define BATCH  4096
#define NOUT   3
#define ROWS   (BATCH * DLEN)      // 409600 (b*100+t flattened, contiguous)
#define TILES  (ROWS / 16)         // 25600
#define WAVES_PER_BLOCK 8

typedef __attribute__((ext_vector_type(2))) float v2f;
typedef __attribute__((ext_vector_type(4))) float v4f;
typedef __attribute__((ext_vector_type(8))) float v8f;

// ---------------------------------------------------------------------------
// Phase 1: cur_ws[o*ROWS + row] = sum_k spike[row][k] * W[o][k]
// One wave per 16-row tile. Spikes staged in LDS (stride 100 dwords is
// bank-conflict-free for the A-fragment ds_load_b64 pattern).
// ---------------------------------------------------------------------------
__global__ __launch_bounds__(256) void snn_phase1_wmma(
    const float* __restrict__ x, const float* __restrict__ u,
    const float* __restrict__ W, float* __restrict__ cur_ws)
{
    __shared__ float spk_lds[WAVES_PER_BLOCK * 16 * DLEN];   // 8*6400B = 51.2KB

    const int lane = threadIdx.x & 31;
    const int wave = threadIdx.x >> 5;
    const int tile = blockIdx.x * WAVES_PER_BLOCK + wave;

    const float* xt = x + (size_t)tile * (16 * DLEN);
    const float* ut = u + (size_t)tile * (16 * DLEN);
    float* s = spk_lds + wave * (16 * DLEN);

    // --- load x,u (coalesced), compute spikes, stage into LDS ---
    // 1600 floats per tile = 12 * (32 lanes * float4) + 32 lanes * float2
#pragma unroll
    for (int i = 0; i < 12; ++i) {
        const int off = i * 128 + lane * 4;
        v4f xv = *(const v4f*)(xt + off);
        v4f uv = *(const v4f*)(ut + off);
        v4f sv;
        sv.x = (uv.x < xv.x) ? 1.0f : 0.0f;
        sv.y = (uv.y < xv.y) ? 1.0f : 0.0f;
        sv.z = (uv.z < xv.z) ? 1.0f : 0.0f;
        sv.w = (uv.w < xv.w) ? 1.0f : 0.0f;
        *(v4f*)(s + off) = sv;
    }
    {
        const int off = 1536 + lane * 2;
        v2f xv = *(const v2f*)(xt + off);
        v2f uv = *(const v2f*)(ut + off);
        v2f sv;
        sv.x = (uv.x < xv.x) ? 1.0f : 0.0f;
        sv.y = (uv.y < xv.y) ? 1.0f : 0.0f;
        *(v2f*)(s + off) = sv;
    }

    // --- B-matrix fragments: B[k][n] = (n<3) ? W[n][k] : 0, K in chunks of 4.
    // Assumed B 4x16 f32 layout (mirror of documented A 16x4):
    //   vgpr0: lanes0-15 = K0, lanes16-31 = K2 ; vgpr1: K1 / K3.
    const int n    = lane & 15;
    const int half = lane >> 4;
    v2f bfrag[25];
    if (n < NOUT) {
#pragma unroll
        for (int kc = 0; kc < 25; ++kc)
            bfrag[kc] = *(const v2f*)(W + n * DLEN + kc * 4 + half * 2);
    } else {
#pragma unroll
        for (int kc = 0; kc < 25; ++kc) { bfrag[kc].x = 0.0f; bfrag[kc].y = 0.0f; }
    }

    __syncthreads();

    // --- A fragments from LDS + 25 chained fp32 WMMAs ---
    // A 16x4 f32 layout: lane L<16 -> row L, K={k0,k1}; lane L+16 -> K={k2,k3}.
    const float* arow = s + (lane & 15) * DLEN + half * 2;
    v8f c = {};
#pragma unroll
    for (int kc = 0; kc < 25; ++kc) {
        v2f a = *(const v2f*)(arow + kc * 4);
        c = __builtin_amdgcn_wmma_f32_16x16x4_f32(
                /*neg_a=*/false, a, /*neg_b=*/false, bfrag[kc],
                /*c_mod=*/(short)0, c, /*reuse_a=*/false, /*reuse_b=*/false);
    }

    // --- store: C/D layout -> lane holds column n for rows (half*8 .. half*8+7)
    if (n < NOUT) {
        float* dst = cur_ws + (size_t)n * ROWS + (size_t)tile * 16 + half * 8;
        v4f lo = { c[0], c[1], c[2], c[3] };
        v4f hi = { c[4], c[5], c[6], c[7] };
        *(v4f*)(dst)     = lo;
        *(v4f*)(dst + 4) = hi;
    }
}

// ---------------------------------------------------------------------------
// Phase 2: leaky integrate-and-fire scan. tid = b*3 + o -> coalesced writes.
// cur_ws is [o][b*100+t] so each thread reads 100 contiguous floats.
// ---------------------------------------------------------------------------
__global__ __launch_bounds__(256) void snn_phase2_scan(
    const float* __restrict__ cur_ws, const float* __restrict__ bias,
    const float* __restrict__ beta_p, const float* __restrict__ thr_p,
    float* __restrict__ out)
{
    const int tid = blockIdx.x * 256 + threadIdx.x;
    if (tid >= BATCH * NOUT) return;
    const int o = tid % NOUT;
    const int b = tid / NOUT;

    const float bias_o = bias[o];
    const float beta   = beta_p[0];
    const float thr    = thr_p[0];

    const float* cp = cur_ws + (size_t)o * ROWS + (size_t)b * DLEN;

    float cur[DLEN];
#pragma unroll
    for (int i = 0; i < DLEN / 4; ++i) {
        v4f v = *(const v4f*)(cp + i * 4);
        cur[i * 4 + 0] = v.x;
        cur[i * 4 + 1] = v.y;
        cur[i * 4 + 2] = v.z;
        cur[i * 4 + 3] = v.w;
    }

    float* spk_out = out;                          // [t][b][o] = t*12288 + tid
    float* mem_out = out + (size_t)ROWS * NOUT;    // second output, same layout

    float mem = 0.0f;
#pragma unroll
    for (int t = 0; t < DLEN; ++t) {
        const float cv    = cur[t] + bias_o;
        const float reset = (mem - thr) > 0.0f ? 1.0f : 0.0f;  // from PREVIOUS mem
        mem = beta * mem + cv - reset * thr;
        const float spk = (mem - thr) > 0.0f ? 1.0f : 0.0f;
        spk_out[(size_t)t * (BATCH * NOUT) + tid] = spk;
        mem_out[(size_t)t * (BATCH * NOUT) + tid] = mem;
    }
}

// ---------------------------------------------------------------------------
extern "C" void kernel_launch(void* const* d_in, const int* in_sizes, int n_in,
                              void* d_out, int out_size, void* d_ws, size_t ws_size,
                              hipStream_t stream) {
    const float* x    = (const float*)d_in[0];   // [4096,100,100]
    const float* u    = (const float*)d_in[1];   // [4096,100,100]
    const float* W    = (const float*)d_in[2];   // [3,100]
    const float* bias = (const float*)d_in[3];   // [3]
    const float* beta = (const float*)d_in[4];   // scalar
    const float* thr  = (const float*)d_in[5];   // scalar

    float* cur_ws = (float*)d_ws;                // 3 * 409600 floats = 4.9 MB

    snn_phase1_wmma<<<TILES / WAVES_PER_BLOCK, 256, 0, stream>>>(x, u, W, cur_ws);
    snn_phase2_scan<<<(BATCH * NOUT + 255) / 256, 256, 0, stream>>>(
        cur_ws, bias, beta, thr, (float*)d_out);
}